// CustomModel_88862873354402
// MI455X (gfx1250) — compile-verified
//
#include <hip/hip_runtime.h>

// Problem constants (from reference)
#define B_   32
#define N_   65536
#define D_   16
#define H_   64
#define M_   32
#define OUT_ 3

typedef float v2f __attribute__((ext_vector_type(2)));
typedef float v8f __attribute__((ext_vector_type(8)));

// D = A(16x4) * B(4x16) + C(16x16), fp32, wave32.
// A layout: lane m(<16): {K0,K1}; lane 16+m: {K2,K3}. B mirrors with N=lane%16.
// C/D layout: VGPR r -> rows {r | r+8} across lane halves, col = lane%16.
__device__ __forceinline__ v8f wmma_f32(v2f a, v2f b, v8f c) {
  return __builtin_amdgcn_wmma_f32_16x16x4_f32(false, a, false, b, (short)0, c,
                                               false, false);
}

// ---------------------------------------------------------------------------
// Kernel 1: hsum[b][h] = sum_n relu(x_n @ mw1 + mb1)[h]   (+ compact x_last)
// grid = (N/2048, B), block = 256 (8 waves, 16 node-groups of 16 per wave)
// ---------------------------------------------------------------------------
__global__ __launch_bounds__(256) void k1_hsum(
    const float* __restrict__ inputs, const float* __restrict__ mw1,
    const float* __restrict__ mb1, float* __restrict__ hsum,
    float* __restrict__ xlast, int write_xlast) {
  const int lane = threadIdx.x & 31;
  const int wave = threadIdx.x >> 5;
  const int nn   = lane & 15;   // col (B/C) or row (A) index
  const int half = lane >> 4;   // lane-half selects K/row offset
  const int b     = blockIdx.y;
  const int node0 = blockIdx.x * 2048 + wave * 256;
  const float* Xb = inputs + (size_t)b * N_ * D_;

  // Preload mw1 [16 x 64] as B-fragments: 4 col-tiles x 4 k-steps, in VGPRs.
  v2f bw1[4][4];
#pragma unroll
  for (int c = 0; c < 4; ++c)
#pragma unroll
    for (int s = 0; s < 4; ++s) {
      const int k   = 4 * s + 2 * half;
      const int col = 16 * c + nn;
      bw1[c][s].x = mw1[(k + 0) * H_ + col];
      bw1[c][s].y = mw1[(k + 1) * H_ + col];
    }
  float bias1[4];
#pragma unroll
  for (int c = 0; c < 4; ++c) bias1[c] = mb1[16 * c + nn];

  v8f sacc[4];
#pragma unroll
  for (int c = 0; c < 4; ++c) sacc[c] = (v8f){0, 0, 0, 0, 0, 0, 0, 0};

  for (int g = 0; g < 16; ++g) {
    const int base  = node0 + g * 16;
    const float* Xt = Xb + (size_t)base * D_;
    // A-fragments for X tile [16 nodes x 16 features], K = D = 16.
    v2f a[4];
#pragma unroll
    for (int s = 0; s < 4; ++s) {
      const int k = 4 * s + 2 * half;
      a[s] = *(const v2f*)(Xt + nn * D_ + k);  // 8B-aligned b64 load
    }
    // Lane half 1, frag 3 holds X[m][14],X[m][15] -> compact x_last.
    if (write_xlast && half)
      xlast[(size_t)b * N_ + base + nn] = a[3].y;

#pragma unroll
    for (int c = 0; c < 4; ++c) {
      v8f h = (v8f){0, 0, 0, 0, 0, 0, 0, 0};
#pragma unroll
      for (int s = 0; s < 4; ++s) h = wmma_f32(a[s], bw1[c][s], h);
#pragma unroll
      for (int r = 0; r < 8; ++r)
        sacc[c][r] += fmaxf(h[r] + bias1[c], 0.0f);
    }
  }

  // Column sums: 8 register rows + cross-half shuffle, then one atomic/lane.
#pragma unroll
  for (int c = 0; c < 4; ++c) {
    float t = 0.0f;
#pragma unroll
    for (int r = 0; r < 8; ++r) t += sacc[c][r];
    t += __shfl_xor(t, 16);
    if (!half) atomicAdd(&hsum[b * H_ + 16 * c + nn], t);
  }
}

// ---------------------------------------------------------------------------
// Kernel 2: out = relu(cvec + x_last*wrow) @ iw2 + ib2, per node.
// cvec = (hsum@mw2 + N*mb2) @ iw1[:32] + ib1,  wrow = iw1[32].
// grid = (N/2048, B), block = 256
// ---------------------------------------------------------------------------
__global__ __launch_bounds__(256) void k2_out(
    const float* __restrict__ inputs, const float* __restrict__ hsum,
    const float* __restrict__ mw2, const float* __restrict__ mb2,
    const float* __restrict__ iw1, const float* __restrict__ ib1,
    const float* __restrict__ iw2, const float* __restrict__ ib2,
    const float* __restrict__ xlast, int use_xlast,
    float* __restrict__ out) {
  __shared__ float s_msg[M_];
  __shared__ float s_cvec[H_];
  __shared__ float s_wrow[H_];
  __shared__ float s_iw2[H_ * OUT_];

  const int t    = threadIdx.x;
  const int lane = t & 31;
  const int wave = t >> 5;
  const int nn   = lane & 15;
  const int half = lane >> 4;
  const int b     = blockIdx.y;
  const int node0 = blockIdx.x * 2048 + wave * 256;

  if (t < M_) {
    float acc = (float)N_ * mb2[t];
    for (int h = 0; h < H_; ++h) acc += hsum[b * H_ + h] * mw2[h * M_ + t];
    s_msg[t] = acc;
  }
  if (t < H_ * OUT_) s_iw2[t] = iw2[t];
  __syncthreads();
  if (t < H_) {
    float acc = ib1[t];
    for (int m = 0; m < M_; ++m) acc += s_msg[m] * iw1[m * H_ + t];
    s_cvec[t] = acc;
    s_wrow[t] = iw1[M_ * H_ + t];
  }
  __syncthreads();

  // iw2 [64 x 3] as zero-padded [64 x 16] B-fragments (16 k-steps), in VGPRs.
  v2f bw2[16];
#pragma unroll
  for (int s = 0; s < 16; ++s) {
    const int k = 4 * s + 2 * half;
    bw2[s].x = (nn < OUT_) ? s_iw2[(k + 0) * OUT_ + nn] : 0.0f;
    bw2[s].y = (nn < OUT_) ? s_iw2[(k + 1) * OUT_ + nn] : 0.0f;
  }
  const float outb = (nn < OUT_) ? ib2[nn] : 0.0f;

  for (int g = 0; g < 16; ++g) {
    const int base = node0 + g * 16;
    float xl;
    if (use_xlast)
      xl = xlast[(size_t)b * N_ + base + nn];
    else
      xl = inputs[((size_t)b * N_ + base + nn) * D_ + (D_ - 1)];

    v8f d = (v8f){0, 0, 0, 0, 0, 0, 0, 0};
#pragma unroll
    for (int s = 0; s < 16; ++s) {
      const int k = 4 * s + 2 * half;
      v2f a;  // g[m][k] built on the fly in A-fragment layout
      a.x = fmaxf(s_cvec[k + 0] + xl * s_wrow[k + 0], 0.0f);
      a.y = fmaxf(s_cvec[k + 1] + xl * s_wrow[k + 1], 0.0f);
      d = wmma_f32(a, bw2[s], d);
    }
    if (nn < OUT_) {
#pragma unroll
      for (int r = 0; r < 8; ++r) {
        const int node = base + r + 8 * half;
        out[((size_t)b * N_ + node) * OUT_ + nn] = d[r] + outb;
      }
    }
  }
}

// ---------------------------------------------------------------------------
extern "C" void kernel_launch(void* const* d_in, const int* in_sizes, int n_in,
                              void* d_out, int out_size, void* d_ws,
                              size_t ws_size, hipStream_t stream) {
  const float* inputs = (const float*)d_in[0];
  const float* mw1    = (const float*)d_in[1];
  const float* mb1    = (const float*)d_in[2];
  const float* mw2    = (const float*)d_in[3];
  const float* mb2    = (const float*)d_in[4];
  const float* iw1    = (const float*)d_in[5];
  const float* ib1    = (const float*)d_in[6];
  const float* iw2    = (const float*)d_in[7];
  const float* ib2    = (const float*)d_in[8];
  float* out = (float*)d_out;

  float* hsum  = (float*)d_ws;               // B*H accumulators
  float* xlast = hsum + B_ * H_;             // compact x_last, B*N floats
  const size_t need =
      (size_t)B_ * H_ * sizeof(float) + (size_t)B_ * N_ * sizeof(float);
  const int use_xlast = (ws_size >= need) ? 1 : 0;

  hipMemsetAsync(d_ws, 0, (size_t)B_ * H_ * sizeof(float), stream);

  dim3 grid(N_ / 2048, B_);
  k1_hsum<<<grid, 256, 0, stream>>>(inputs, mw1, mb1, hsum, xlast, use_xlast);
  k2_out<<<grid, 256, 0, stream>>>(inputs, hsum, mw2, mb2, iw1, ib1, iw2, ib2,
                                   xlast, use_xlast, out);
}